// chebyshevConvolution_43559558316210
// MI455X (gfx1250) — compile-verified
//
#include <hip/hip_runtime.h>

#define M_NODES 20000
#define N_BATCH 16
#define FIN 64
#define FOUT 64
#define KCH 6
#define NNZ 160000
#define FW (FIN * N_BATCH) /* 1024 features per graph-node row */

typedef __attribute__((ext_vector_type(2))) float v2f;
typedef __attribute__((ext_vector_type(8))) float v8f;

// T0[m*1024 + n*64 + f] = x[n][m][f]   (term layout: [M][N][FIN], contiguous in f)
__global__ void cheb_permute_kernel(const float* __restrict__ x, float* __restrict__ T0) {
  int idx = blockIdx.x * blockDim.x + threadIdx.x; // < 20,480,000
  int m = idx >> 10;
  int rem = idx & 1023;
  int n = rem >> 6;
  int f = rem & 63;
  T0[idx] = x[((size_t)n * M_NODES + m) * FIN + f];
}

__global__ void cheb_init_bias_kernel(float* __restrict__ out, const float* __restrict__ bias) {
  int idx = blockIdx.x * blockDim.x + threadIdx.x;
  out[idx] = bias[idx & 63];
}

__global__ void cheb_zero_kernel(float* __restrict__ Y) {
  int idx = blockIdx.x * blockDim.x + threadIdx.x;
  Y[idx] = 0.0f;
}

__global__ void cheb_negcopy_kernel(float* __restrict__ Y, const float* __restrict__ A) {
  int idx = blockIdx.x * blockDim.x + threadIdx.x;
  Y[idx] = -A[idx];
}

// Y[row] += scale * val * X[col]  over 1024 features; one block per nnz edge.
__global__ void cheb_spmm_scatter_kernel(float* __restrict__ Y, const float* __restrict__ X,
                                         const float* __restrict__ vals,
                                         const int* __restrict__ rows,
                                         const int* __restrict__ cols, float scale) {
  int e = blockIdx.x;
  int j = threadIdx.x;
  int row = rows[e];
  int col = cols[e];
  float v = vals[e] * scale;
  const float* xp = X + (size_t)col * FW;
  float* yp = Y + (size_t)row * FW;
#pragma unroll
  for (int t = 0; t < 4; ++t) {
    int f = j + t * 256;
    unsafeAtomicAdd(yp + f, v * xp[f]);
  }
}

// out[(n*M+m), fout] += sum_f T[m][n][f] * weight[(f*K + kterm)*FOUT + fout]
// One wave per 16x16 tile; V_WMMA_F32_16X16X4_F32, K-loop of 64 in steps of 4.
__global__ __launch_bounds__(256) void cheb_gemm_acc_kernel(
    float* __restrict__ out, const float* __restrict__ T,
    const float* __restrict__ W, int kterm) {
  int wave = blockIdx.x * 8 + (threadIdx.x >> 5);
  int lane = threadIdx.x & 31;
  int col_tile = wave & 3;           // 0..3   (FOUT / 16)
  int row_tile = wave >> 2;          // 0..19999 = n*1250 + m_base/16
  int n = row_tile / 1250;
  int m_base = (row_tile - n * 1250) << 4;

  int r = lane & 15;                 // row within tile (A) / col within tile (B,C,D)
  int half = lane >> 4;              // K-half selector for A/B fragments
  int fout = (col_tile << 4) + r;

  // A: lane holds row (m_base + r), two consecutive f values per VGPR pair.
  const float* Trow = T + ((size_t)(m_base + r) << 10) + ((size_t)n << 6);
  // B: lane holds column fout; f strides by KCH*FOUT in weight.
  const float* Wcol = W + (size_t)kterm * FOUT + fout;

  // C/D: VGPR v holds rows (v + half*8), col fout.
  size_t out_base = (((size_t)n * M_NODES + m_base + half * 8) << 6) + fout;

  v8f c;
#pragma unroll
  for (int v = 0; v < 8; ++v)
    c[v] = out[out_base + ((size_t)v << 6)];

#pragma unroll
  for (int kk = 0; kk < FIN; kk += 4) {
    int fA = kk + half * 2;
    v2f a = *(const v2f*)(Trow + fA);          // 8B-aligned: even float offset
    v2f b;
    b.x = Wcol[(size_t)(fA + 0) * (KCH * FOUT)];
    b.y = Wcol[(size_t)(fA + 1) * (KCH * FOUT)];
    c = __builtin_amdgcn_wmma_f32_16x16x4_f32(
        /*neg_a=*/false, a, /*neg_b=*/false, b,
        /*c_mod=*/(short)0, c, /*reuse_a=*/false, /*reuse_b=*/false);
  }

#pragma unroll
  for (int v = 0; v < 8; ++v)
    out[out_base + ((size_t)v << 6)] = c[v];
}

extern "C" void kernel_launch(void* const* d_in, const int* in_sizes, int n_in,
                              void* d_out, int out_size, void* d_ws, size_t ws_size,
                              hipStream_t stream) {
  const float* x    = (const float*)d_in[0];
  const float* Lv   = (const float*)d_in[1];
  const float* W    = (const float*)d_in[2];
  const float* bias = (const float*)d_in[3];
  const int*   Lrow = (const int*)d_in[4];
  const int*   Lcol = (const int*)d_in[5];
  float* out = (float*)d_out;

  const size_t S = (size_t)M_NODES * FW; // 20,480,000 floats per term buffer
  float* buf0 = (float*)d_ws;
  float* buf1 = buf0 + S;
  float* buf2 = buf1 + S;
  float* buf[3] = {buf0, buf1, buf2};

  dim3 blk(256);
  dim3 grid_elt((unsigned)(S / 256));  // 80,000 blocks, covers out too (same size)
  dim3 grid_gemm(10000);               // 80,000 waves = 20000 row-tiles x 4 col-tiles

  // x0 (term 0) and bias-seeded output
  cheb_permute_kernel<<<grid_elt, blk, 0, stream>>>(x, buf[0]);
  cheb_init_bias_kernel<<<grid_elt, blk, 0, stream>>>(out, bias);
  cheb_gemm_acc_kernel<<<grid_gemm, blk, 0, stream>>>(out, buf[0], W, 0);

  // term 1: x1 = L x0
  cheb_zero_kernel<<<grid_elt, blk, 0, stream>>>(buf[1]);
  cheb_spmm_scatter_kernel<<<NNZ, blk, 0, stream>>>(buf[1], buf[0], Lv, Lrow, Lcol, 1.0f);
  cheb_gemm_acc_kernel<<<grid_gemm, blk, 0, stream>>>(out, buf[1], W, 1);

  // terms 2..5: xk = 2 L x_{k-1} - x_{k-2}
  int km2 = 0, km1 = 1, kc = 2;
  for (int k = 2; k < KCH; ++k) {
    cheb_negcopy_kernel<<<grid_elt, blk, 0, stream>>>(buf[kc], buf[km2]);
    cheb_spmm_scatter_kernel<<<NNZ, blk, 0, stream>>>(buf[kc], buf[km1], Lv, Lrow, Lcol, 2.0f);
    cheb_gemm_acc_kernel<<<grid_gemm, blk, 0, stream>>>(out, buf[kc], W, k);
    int t = km2; km2 = km1; km1 = kc; kc = t;
  }

  (void)in_sizes; (void)n_in; (void)out_size; (void)ws_size;
}